// MVLoss_19121194402254
// MI455X (gfx1250) — compile-verified
//
#include <hip/hip_runtime.h>

typedef float v2f __attribute__((ext_vector_type(2)));
typedef float v8f __attribute__((ext_vector_type(8)));

// One-sided Chamfer distance via V_WMMA_F32_16X16X4_F32.
// Each wave: 16 source rows, scans all P targets in 16-wide tiles.
// The 4th K-slot carries the ||t||^2 term:
//   A = [-2*s | 1],  B = [t ; ||t||^2],  C = 0
//   D[m][n] = ||t_n||^2 - 2 s_m . t_n
// min over n, then add ||s_m||^2 at the end.
__global__ __launch_bounds__(256) void chamfer_wmma_f32(
    const float* __restrict__ src,   // (B, P, 3)
    const float* __restrict__ tgt,   // (B, P, 3)
    float* __restrict__ out,         // scalar
    int B, int P)
{
    const int lane        = threadIdx.x & 31;
    const int waveInBlock = threadIdx.x >> 5;
    const int wave        = blockIdx.x * (blockDim.x >> 5) + waveInBlock;
    const int wavesPerB   = P >> 4;           // 16 source rows per wave
    const int b           = wave / wavesPerB; // uniform across the wave
    if (b >= B) return;
    const int rowBase = (wave % wavesPerB) << 4;

    const int half = lane >> 4;   // 0: lanes 0-15 (K=0,1), 1: lanes 16-31 (K=2,3)
    const int sub  = lane & 15;

    // ---- Load this wave's 16 source rows; build A = [-2*S | 1] (16x4) ----
    const float* sp = src + ((size_t)b * P + (rowBase + sub)) * 3;
    const float sx = sp[0], sy = sp[1], sz = sp[2];
    // per-lane partial of ||s_row||^2: halves hold (x^2+y^2) and (z^2)
    const float ns_partial = half ? (sz * sz) : (sx * sx + sy * sy);

    v2f a;
    a.x = half ? (-2.0f * sz) : (-2.0f * sx);  // K=2 : K=0
    a.y = half ? 1.0f         : (-2.0f * sy);  // K=3 : K=1

    v8f czero;
#pragma unroll
    for (int k = 0; k < 8; ++k) czero[k] = 0.0f;   // loop-invariant C

    v8f minv;
#pragma unroll
    for (int k = 0; k < 8; ++k) minv[k] = 3.0e38f;

    const float* tb = tgt + (size_t)b * P * 3;

    // ---- Scan all targets, 16 per WMMA ----
    for (int j0 = 0; j0 < P; j0 += 16) {
        const float* tp = tb + (size_t)(j0 + sub) * 3;
        const float tx = tp[0], ty = tp[1], tz = tp[2];
        const float nt = tx * tx + ty * ty + tz * tz;  // ||t_N||^2, N = sub

        v2f bm;                       // B matrix 4x16, column N = sub
        bm.x = half ? tz : tx;        // K=2 : K=0
        bm.y = half ? nt : ty;        // K=3 : K=1  (K=3 row carries ||t||^2)

        // D = A*B + 0  ->  ||t||^2 - 2 s.t
        v8f d = __builtin_amdgcn_wmma_f32_16x16x4_f32(
            false, a, false, bm, (short)0, czero, false, false);

#pragma unroll
        for (int k = 0; k < 8; ++k) minv[k] = fminf(minv[k], d[k]);
    }

    // ---- Min-reduce across the 16 lanes of each half (rows v / v+8) ----
#pragma unroll
    for (int off = 1; off <= 8; off <<= 1) {
#pragma unroll
        for (int k = 0; k < 8; ++k) {
            float o = __shfl_xor(minv[k], off, 32);
            minv[k] = fminf(minv[k], o);
        }
    }
    float min_sum = 0.0f;
#pragma unroll
    for (int k = 0; k < 8; ++k) min_sum += minv[k];   // uniform within each half
    min_sum += __shfl_xor(min_sum, 16, 32);           // rows 0-7 + rows 8-15

    // ---- Sum of ||s_row||^2 over the wave's 16 rows (full-wave reduce) ----
    float ns_sum = ns_partial;
#pragma unroll
    for (int off = 1; off <= 16; off <<= 1)
        ns_sum += __shfl_xor(ns_sum, off, 32);

    if (lane == 0) {
        atomicAdd(out, (min_sum + ns_sum) * (1.0f / (float)B));
    }
}

extern "C" void kernel_launch(void* const* d_in, const int* in_sizes, int n_in,
                              void* d_out, int out_size, void* d_ws, size_t ws_size,
                              hipStream_t stream) {
    const float* src = (const float*)d_in[0];   // (B, P, 3) float32
    const float* tgt = (const float*)d_in[1];   // (B, P, 3) float32
    // d_in[2] is the Cayley table; unused (geometric-product scalar == squared norm)
    float* out = (float*)d_out;

    const int B = 4;
    const int P = in_sizes[0] / (B * 3);        // 8192

    hipMemsetAsync(out, 0, sizeof(float), stream);

    const int wavesTotal    = (B * P) / 16;     // one wave per 16 source rows
    const int wavesPerBlock = 8;                // 256 threads
    const int blocks        = wavesTotal / wavesPerBlock;
    chamfer_wmma_f32<<<blocks, 256, 0, stream>>>(src, tgt, out, B, P);
}